// ProjectRenderScoreModel_69252052680736
// MI455X (gfx1250) — compile-verified
//
#include <hip/hip_runtime.h>
#include <hip/hip_bf16.h>

typedef __attribute__((ext_vector_type(2))) float v2f;
typedef __attribute__((ext_vector_type(8))) float v8f;

#define BS   2
#define NCAM 3
#define NPT  64
#define IMS  256
#define NIMG (BS * NCAM)          // 6
#define PIX  (IMS * IMS)          // 65536
#define LOG2E 1.44269504088896340736f

// Native gfx1250 trans-pipe ops: v_log_f32 (returns log2) and v_exp_f32 (exp2).
__device__ __forceinline__ float fast_log2(float x) { return __builtin_amdgcn_logf(x); }
__device__ __forceinline__ float fast_exp2(float x) { return __builtin_amdgcn_exp2f(x); }

// One block = one image (bc) x 2 rows. 8 waves; each wave = one quarter-row
// (64 pixels). Per wave: 4 pixel tiles x 4 point tiles of V_WMMA_F32_16X16X4_F32
// computing the squared-distance field, then the exp chain on the trans pipe.
__global__ __launch_bounds__(256) void render_blobs_wmma(
    const float* __restrict__ points,        // (BS,NCAM,N,2)
    const float* __restrict__ sigmas,        // (BS,N)
    const float* __restrict__ exponents,     // (BS,N)
    const float* __restrict__ intensities,   // (BS,N)
    const float* __restrict__ cam_sig,       // (BS,NCAM)
    const float* __restrict__ cam_exp,       // (BS,NCAM)
    const float* __restrict__ cam_int,       // (BS,NCAM)
    float* __restrict__ masks,               // (BS,NCAM,H,W)
    float* __restrict__ blobs)               // (BS,NCAM,N,H,W)
{
    __shared__ float s_px[NPT], s_py[NPT], s_inv[NPT], s_e[NPT], s_it[NPT];

    const int bc     = blockIdx.x >> 7;      // image index 0..5 (128 blocks/img)
    const int rowblk = blockIdx.x & 127;     // 2-row strip within image
    const int b      = bc / NCAM;
    const int cam    = bc - b * NCAM;
    const int tid    = threadIdx.x;

    // Stage per-point parameters in LDS (one image's worth).
    if (tid < NPT) {
        const int n = tid;
        float pxr = points[((size_t)bc * NPT + n) * 2 + 0];
        float pyr = points[((size_t)bc * NPT + n) * 2 + 1];
        // normalize to [-1,1] with clip
        float px = fminf(fmaxf((pxr - 128.0f) * (1.0f / 128.0f), -1.0f), 1.0f);
        float py = fminf(fmaxf((pyr - 128.0f) * (1.0f / 128.0f), -1.0f), 1.0f);
        float sg = sigmas[b * NPT + n] * cam_sig[b * NCAM + cam];
        s_px[n]  = px;
        s_py[n]  = py;
        s_inv[n] = 0.5f / (sg * sg);                               // 1/(2 sig^2)
        s_e[n]   = exponents[b * NPT + n]   * cam_exp[b * NCAM + cam];
        s_it[n]  = intensities[b * NPT + n] * cam_int[b * NCAM + cam];
    }
    __syncthreads();

    const int wave  = tid >> 5;
    const int lane  = tid & 31;
    const int laneM = lane & 15;             // matrix row/col index within tile
    const int hi    = lane >> 4;             // lane half selects K pair / M+8
    const int y     = rowblk * 2 + (wave >> 2);
    const int x0    = (wave & 3) * 64;       // quarter-row base
    const float gy  = (float)y * (2.0f / 255.0f) - 1.0f;

    // A matrix (16 points x K=4) for each of the 4 point tiles.
    // Lanes 0-15: VGPR0=K0 (px^2+py^2), VGPR1=K1 (-2px) for point m=laneM.
    // Lanes 16-31: VGPR0=K2 (-2py),     VGPR1=K3 (1).
    v2f a[4];
#pragma unroll
    for (int pt = 0; pt < 4; ++pt) {
        const int m = pt * 16 + laneM;
        const float px = s_px[m], py = s_py[m];
        a[pt].x = hi ? (-2.0f * py) : (px * px + py * py);
        a[pt].y = hi ? 1.0f         : (-2.0f * px);
    }

    float* __restrict__ blobrow = blobs + (size_t)bc * NPT * PIX + (size_t)y * IMS;

#pragma unroll
    for (int t = 0; t < 4; ++t) {
        const int xb = x0 + t * 16;
        const float gx = (float)(xb + laneM) * (2.0f / 255.0f) - 1.0f;
        // B matrix (K=4 x 16 pixels), mirror of A layout:
        // lanes 0-15: VGPR0=K0 (1), VGPR1=K1 (gx); lanes 16-31: K2 (gy), K3 (gx^2+gy^2)
        v2f bv;
        bv.x = hi ? gy : 1.0f;
        bv.y = hi ? (gx * gx + gy * gy) : gx;

        float mv = 0.0f;                     // running per-pixel max (this half)
#pragma unroll
        for (int pt = 0; pt < 4; ++pt) {
            v8f c = {};                      // D = A x B (+0)
            c = __builtin_amdgcn_wmma_f32_16x16x4_f32(
                    /*neg_a=*/false, a[pt], /*neg_b=*/false, bv,
                    /*c_mod=*/(short)0, c, /*reuse_a=*/false, /*reuse_b=*/false);

            const int mbase = pt * 16 + hi * 8;
#pragma unroll
            for (int r = 0; r < 8; ++r) {
                const int m = mbase + r;     // same for all 16 lanes of a half -> LDS broadcast
                const float dst = c[r];
                // blob = exp(-(dst/(2sig^2))^e) = exp2(-log2e * exp2(e*log2(t)))
                const float tt = dst * s_inv[m];
                const float u  = fast_exp2(s_e[m] * fast_log2(tt));
                const float blob = fast_exp2(-LOG2E * u);
                blobrow[(size_t)m * PIX + xb + laneM] = blob;
                mv = fmaxf(mv, blob * s_it[m]);
            }
        }
        // combine lane halves (rows 0-7 vs 8-15 of each point tile live in
        // opposite halves): wave32 xor-shuffle across lane 16.
        const float other = __shfl_xor(mv, 16, 32);
        mv = fmaxf(mv, other);
        if (lane < 16)
            masks[(size_t)bc * PIX + (size_t)y * IMS + xb + lane] = fminf(mv, 1.0f);
    }
}

extern "C" void kernel_launch(void* const* d_in, const int* in_sizes, int n_in,
                              void* d_out, int out_size, void* d_ws, size_t ws_size,
                              hipStream_t stream) {
    (void)in_sizes; (void)n_in; (void)out_size; (void)d_ws; (void)ws_size;
    const float* points      = (const float*)d_in[0];
    const float* sigmas      = (const float*)d_in[1];
    const float* exponents   = (const float*)d_in[2];
    const float* intensities = (const float*)d_in[3];
    const float* cam_sig     = (const float*)d_in[4];
    const float* cam_exp     = (const float*)d_in[5];
    const float* cam_int     = (const float*)d_in[6];
    // d_in[7] = image_size (compile-time 256)

    float* masks = (float*)d_out;                     // (2,3,256,256)
    float* blobs = masks + (size_t)NIMG * PIX;        // (2,3,64,256,256)

    dim3 grid(NIMG * 128);   // 6 images x 128 two-row strips
    dim3 block(256);         // 8 waves (wave32)
    render_blobs_wmma<<<grid, block, 0, stream>>>(
        points, sigmas, exponents, intensities,
        cam_sig, cam_exp, cam_int, masks, blobs);
}